// GIN_2903397892177
// MI455X (gfx1250) — compile-verified
//
#include <hip/hip_runtime.h>
#include <hip/hip_bf16.h>

typedef __attribute__((ext_vector_type(2))) float v2f;
typedef __attribute__((ext_vector_type(8))) float v8f;

#define N_NODES_C 100000
#define N_EDGES_C 1600000
#define FEAT 128
#define N_CLS 10
#define N_LAYERS 5
#define N_GRAPHS 512
#define BN_EPS_C 1e-5f
#define LDS_STRIDE 132  // padded row stride (floats); 132*4=528B keeps 16B align

// ---------------- copy cur -> t (the "x +" term of GIN) and zero BN stats ----
__global__ void copy_zero_stats(const float* __restrict__ src, float* __restrict__ dst,
                                float* __restrict__ stats, long n4) {
  if (blockIdx.x == 0 && threadIdx.x < 256) stats[threadIdx.x] = 0.f;
  const float4* s = (const float4*)src;
  float4* d = (float4*)dst;
  for (long i = (long)blockIdx.x * blockDim.x + threadIdx.x; i < n4;
       i += (long)gridDim.x * blockDim.x)
    d[i] = s[i];
}

// ---------------- edge scatter: t[dst] += x[src], 32 lanes per edge ----------
__global__ void edge_scatter(const float* __restrict__ x, float* t,
                             const int* __restrict__ src, const int* __restrict__ dst) {
  long idx = (long)blockIdx.x * blockDim.x + threadIdx.x;
  long e = idx >> 5;
  int lane = (int)(idx & 31);
  if (e >= N_EDGES_C) return;
  int s = src[e], d = dst[e];
  float4 v = ((const float4*)(x + (long)s * FEAT))[lane];
  float* o = t + (long)d * FEAT + lane * 4;
  atomicAdd(o + 0, v.x);
  atomicAdd(o + 1, v.y);
  atomicAdd(o + 2, v.z);
  atomicAdd(o + 3, v.w);
}

// ---------------- WMMA GEMM: Out[M x 128] = op(A) @ W(128x128) + bias --------
// Block = SUB*16 rows x 128 cols of output; 8 waves, SUB 16x16 tiles each
// (B fragment reused SUB times -> 1/SUB the weight traffic).
// In-place safe (Out may alias A): A panel fully staged in LDS before stores.
// Non-transforming path stages the panel with async global->LDS B128 copies.
template <int SUB, bool PRE_AFF, bool POST_RELU>
__global__ __launch_bounds__(256) void gemm128_wmma(
    const float* A, float* Out, const float* __restrict__ W,
    const float* __restrict__ bias, const float* __restrict__ aff_a,
    const float* __restrict__ aff_c) {
  constexpr int ROWS = SUB * 16;
  __shared__ float As[ROWS * LDS_STRIDE];
  const int tid = threadIdx.x;
  const long r0 = (long)blockIdx.x * ROWS;

  if (PRE_AFF) {
    // stage panel, applying BN-affine + ReLU on the fly
    for (int idx = tid; idx < ROWS * 32; idx += 256) {
      int row = idx >> 5;
      int c4 = (idx & 31) * 4;
      float4 v = *(const float4*)(A + (r0 + row) * FEAT + c4);
      float* l = As + row * LDS_STRIDE + c4;
      float t0 = aff_a[c4 + 0] * v.x + aff_c[c4 + 0];
      float t1 = aff_a[c4 + 1] * v.y + aff_c[c4 + 1];
      float t2 = aff_a[c4 + 2] * v.z + aff_c[c4 + 2];
      float t3 = aff_a[c4 + 3] * v.w + aff_c[c4 + 3];
      l[0] = t0 > 0.f ? t0 : 0.f;
      l[1] = t1 > 0.f ? t1 : 0.f;
      l[2] = t2 > 0.f ? t2 : 0.f;
      l[3] = t3 > 0.f ? t3 : 0.f;
    }
    __syncthreads();
  } else {
    // async global -> LDS staging (ASYNCcnt path), 16B chunks
    for (int idx = tid; idx < ROWS * 32; idx += 256) {
      int row = idx >> 5;
      int c4 = (idx & 31) * 4;
      const float* gp = A + (r0 + row) * FEAT + c4;
      unsigned loff = (unsigned)(uintptr_t)(const void*)(As + row * LDS_STRIDE + c4);
      asm volatile("global_load_async_to_lds_b128 %0, %1, off"
                   :
                   : "v"(loff), "v"(gp)
                   : "memory");
    }
    asm volatile("s_wait_asynccnt 0x0" ::: "memory");
    __syncthreads();
  }

  const int wave = tid >> 5;  // 0..7 -> output column tile
  const int lane = tid & 31;
  const int n0 = wave * 16;
  const int mi = lane & 15;        // A: M index, B: N index
  const int kh = (lane >> 4) * 2;  // 0 or 2 (K half per §7.12.2)

  const v8f vzero = {};
  v8f acc[SUB];
#pragma unroll
  for (int s = 0; s < SUB; ++s) acc[s] = vzero;

#pragma unroll
  for (int k0 = 0; k0 < FEAT; k0 += 4) {
    v2f b;
    const float* wp = W + (long)(k0 + kh) * FEAT + n0 + mi;  // W is KxN row-major
    b.x = wp[0];
    b.y = wp[FEAT];
#pragma unroll
    for (int s = 0; s < SUB; ++s) {
      v2f a;
      const float* ap = As + (s * 16 + mi) * LDS_STRIDE + k0 + kh;
      a.x = ap[0];
      a.y = ap[1];
      acc[s] = __builtin_amdgcn_wmma_f32_16x16x4_f32(false, a, false, b, (short)0,
                                                     acc[s], false, false);
    }
  }

  // C/D layout: VGPR v -> M = v (lanes 0-15) / v+8 (lanes 16-31), N = lane&15
  const int mb = (lane >> 4) * 8;
  const float bv = bias[n0 + mi];
#pragma unroll
  for (int s = 0; s < SUB; ++s) {
    float* o = Out + (r0 + s * 16 + mb) * FEAT + n0 + mi;
#pragma unroll
    for (int v = 0; v < 8; ++v) {
      float val = acc[s][v] + bv;
      if (POST_RELU) val = val > 0.f ? val : 0.f;
      o[(long)v * FEAT] = val;
    }
  }
}

// ---------------- per-feature sum / sumsq over all nodes ---------------------
__global__ void col_stats(const float* __restrict__ h, float* __restrict__ stats) {
  int f = threadIdx.x;  // 128 threads
  float s = 0.f, s2 = 0.f;
  for (long r = blockIdx.x; r < N_NODES_C; r += gridDim.x) {
    float v = h[r * FEAT + f];
    s += v;
    s2 += v * v;
  }
  atomicAdd(&stats[f], s);
  atomicAdd(&stats[128 + f], s2);
}

// ---------------- fold BN into affine: h' = a*h + c --------------------------
__global__ void bn_coef(const float* __restrict__ stats, const float* __restrict__ gamma,
                        const float* __restrict__ beta, float* __restrict__ aff) {
  int f = threadIdx.x;
  const float inv_n = 1.f / (float)N_NODES_C;
  float mu = stats[f] * inv_n;
  float var = stats[128 + f] * inv_n - mu * mu;
  float a = gamma[f] * rsqrtf(var + BN_EPS_C);
  aff[f] = a;
  aff[128 + f] = beta[f] - a * mu;
}

__global__ void zero_f(float* p, int n) {
  int i = blockIdx.x * blockDim.x + threadIdx.x;
  if (i < n) p[i] = 0.f;
}

// ---------------- global add pool: pooled[batch[i]] += x[i] ------------------
__global__ void pool_scatter(const float* __restrict__ x, const int* __restrict__ batch,
                             float* pooled) {
  long idx = (long)blockIdx.x * blockDim.x + threadIdx.x;
  long node = idx >> 5;
  int lane = (int)(idx & 31);
  if (node >= N_NODES_C) return;
  int g = batch[node];
  float4 v = ((const float4*)(x + node * FEAT))[lane];
  float* o = pooled + (long)g * FEAT + lane * 4;
  atomicAdd(o + 0, v.x);
  atomicAdd(o + 1, v.y);
  atomicAdd(o + 2, v.z);
  atomicAdd(o + 3, v.w);
}

// ---------------- head layer 2: 512x128 @ 128x10 + b -------------------------
__global__ void head_out(const float* __restrict__ hidden, const float* __restrict__ W2,
                         const float* __restrict__ b2, float* __restrict__ out) {
  int g = blockIdx.x;   // 512 graphs
  int c = threadIdx.x;  // 32 lanes, first 10 active
  if (c >= N_CLS) return;
  float acc = b2[c];
  for (int k = 0; k < FEAT; ++k) acc += hidden[g * FEAT + k] * W2[k * N_CLS + c];
  out[g * N_CLS + c] = acc;
}

extern "C" void kernel_launch(void* const* d_in, const int* in_sizes, int n_in,
                              void* d_out, int out_size, void* d_ws, size_t ws_size,
                              hipStream_t stream) {
  const float* x_in    = (const float*)d_in[0];
  const int*   ei      = (const int*)d_in[1];
  const int*   batch   = (const int*)d_in[2];
  const float* conv_W1 = (const float*)d_in[3];
  const float* conv_b1 = (const float*)d_in[4];
  const float* conv_g  = (const float*)d_in[5];
  const float* conv_be = (const float*)d_in[6];
  const float* conv_W2 = (const float*)d_in[7];
  const float* conv_b2 = (const float*)d_in[8];
  const float* head_W1 = (const float*)d_in[9];
  const float* head_b1 = (const float*)d_in[10];
  const float* head_W2 = (const float*)d_in[11];
  const float* head_b2 = (const float*)d_in[12];
  const int* src = ei;
  const int* dst = ei + N_EDGES_C;

  float* buf0  = (float*)d_ws;
  float* buf1  = buf0 + (size_t)N_NODES_C * FEAT;
  float* stats = buf1 + (size_t)N_NODES_C * FEAT;
  float* aff   = stats + 256;
  float* pooled = buf1;                    // buf1 is dead after layer 4
  float* hidden = buf1 + (size_t)N_GRAPHS * FEAT;

  const long n4 = (long)N_NODES_C * FEAT / 4;
  const long et = (long)N_EDGES_C * 32;
  const int edge_blocks = (int)((et + 255) / 256);
  const int row_tiles = N_NODES_C / 80;  // SUB=5 -> 80 rows/block -> 1250 blocks

  const float* cur = x_in;
  for (int l = 0; l < N_LAYERS; ++l) {
    float* t = (l & 1) ? buf1 : buf0;
    copy_zero_stats<<<2048, 256, 0, stream>>>(cur, t, stats, n4);
    edge_scatter<<<edge_blocks, 256, 0, stream>>>(cur, t, src, dst);
    gemm128_wmma<5, false, false><<<row_tiles, 256, 0, stream>>>(
        t, t, conv_W1 + (size_t)l * FEAT * FEAT, conv_b1 + l * FEAT, nullptr, nullptr);
    col_stats<<<1024, FEAT, 0, stream>>>(t, stats);
    bn_coef<<<1, FEAT, 0, stream>>>(stats, conv_g + l * FEAT, conv_be + l * FEAT, aff);
    gemm128_wmma<5, true, true><<<row_tiles, 256, 0, stream>>>(
        t, t, conv_W2 + (size_t)l * FEAT * FEAT, conv_b2 + l * FEAT, aff, aff + 128);
    cur = t;  // layer output lives in t
  }

  // cur == buf0 after 5 layers; pooled/hidden overlay buf1 (now dead)
  zero_f<<<(N_GRAPHS * FEAT + 255) / 256, 256, 0, stream>>>(pooled, N_GRAPHS * FEAT);
  const long pt = (long)N_NODES_C * 32;
  pool_scatter<<<(int)((pt + 255) / 256), 256, 0, stream>>>(cur, batch, pooled);

  gemm128_wmma<1, false, true><<<N_GRAPHS / 16, 256, 0, stream>>>(
      pooled, hidden, head_W1, head_b1, nullptr, nullptr);
  head_out<<<N_GRAPHS, 32, 0, stream>>>(hidden, head_W2, head_b2, (float*)d_out);
}